// ChamferDistance_89764816486827
// MI455X (gfx1250) — compile-verified
//
#include <hip/hip_runtime.h>

typedef __attribute__((ext_vector_type(2))) float v2f;
typedef __attribute__((ext_vector_type(8))) float v8f;

// One direction: for each point in A, min squared distance to any point in B.
// Adds (sum_i min_j d(a_i, b_j)) * inv_na to *out.
// Each wave owns one 16-point A tile; loops over all B tiles with one
// V_WMMA_F32_16X16X4_F32 per 16x16 tile of pairs.
__global__ __launch_bounds__(256) void chamfer_min_dir_kernel(
    const float* __restrict__ A,
    const float* __restrict__ B,
    int nb,            // number of B points (multiple of 16)
    float inv_na,      // 1 / number of A points
    float* __restrict__ out)
{
    const int lane = threadIdx.x & 31;
    const int wave = threadIdx.x >> 5;
    const int atile = blockIdx.x * 8 + wave;
    const int col  = lane & 15;
    const bool lo  = lane < 16;

    // ---- Load this wave's 16-point A tile (lane L <-> row L&15) ----
    const float* ap = A + (size_t)(atile * 16 + col) * 3;
    const float ax = ap[0], ay = ap[1], az = ap[2];
    const float asq = fmaf(ax, ax, fmaf(ay, ay, az * az));

    // A operand (16x4 f32, K=3 padded to 4), scaled by -2 so the WMMA
    // directly produces -2 * (a . b).
    // VGPR0: lanes0-15 K=0 (x), lanes16-31 K=2 (z)
    // VGPR1: lanes0-15 K=1 (y), lanes16-31 K=3 (0)
    v2f aop;
    aop.x = -2.0f * (lo ? ax : az);
    aop.y = lo ? (-2.0f * ay) : 0.0f;

    // Running min of (bsq - 2 a.b) per (row, column-lane).
    float m[8];
#pragma unroll
    for (int r = 0; r < 8; ++r) m[r] = __builtin_inff();

    const int nbt = nb >> 4;
    const float* bbase = B + (size_t)col * 3;

#pragma unroll 2
    for (int bt = 0; bt < nbt; ++bt) {
        // ---- Load B tile point for this lane (lane L <-> column L&15) ----
        const float* bp = bbase + (size_t)bt * 48;   // 16 points * 3 floats
        const float bx = bp[0], by = bp[1], bz = bp[2];

        // B operand (4x16 f32): mirror of A layout.
        v2f bop;
        bop.x = lo ? bx : bz;
        bop.y = lo ? by : 0.0f;

        const float bsq = fmaf(bx, bx, fmaf(by, by, bz * bz));

        // C = A(-2a) x B(b): c[r] = -2 * dot(a_row, b_col)
        v8f c = {};
        c = __builtin_amdgcn_wmma_f32_16x16x4_f32(
                /*neg_a=*/false, aop,
                /*neg_b=*/false, bop,
                /*c_mod=*/(short)0, c,
                /*reuse_a=*/false, /*reuse_b=*/false);

        // Fold ||b||^2 (per-lane == per-column) and accumulate running min.
#pragma unroll
        for (int r = 0; r < 8; ++r)
            m[r] = fminf(m[r], c[r] + bsq);
    }

    // ---- Row-min over the 16 column lanes (stays within each 16-lane half;
    // lanes 0-15 hold rows r, lanes 16-31 hold rows r+8). ----
#pragma unroll
    for (int r = 0; r < 8; ++r) {
        float v = m[r];
        v = fminf(v, __shfl_xor(v, 1, 32));
        v = fminf(v, __shfl_xor(v, 2, 32));
        v = fminf(v, __shfl_xor(v, 4, 32));
        v = fminf(v, __shfl_xor(v, 8, 32));
        m[r] = v;
    }

    // lane 0: sum of row-mins for rows 0..7 ; lane 16: rows 8..15
    float partial = ((m[0] + m[1]) + (m[2] + m[3])) +
                    ((m[4] + m[5]) + (m[6] + m[7]));
    partial += __shfl_xor(partial, 16, 32);   // lane 0 now has all 16 rows

    // Sum of ||a||^2 over the 16 rows of this tile (min_j d = asq + min_j(...)).
    float s = asq;
    s += __shfl_xor(s, 1, 32);
    s += __shfl_xor(s, 2, 32);
    s += __shfl_xor(s, 4, 32);
    s += __shfl_xor(s, 8, 32);                // lane 0: sum over its 16 rows

    if (lane == 0) {
        atomicAdd(out, (partial + s) * inv_na);
    }
}

extern "C" void kernel_launch(void* const* d_in, const int* in_sizes, int n_in,
                              void* d_out, int out_size, void* d_ws, size_t ws_size,
                              hipStream_t stream) {
    (void)n_in; (void)d_ws; (void)ws_size; (void)out_size;

    const float* adv = (const float*)d_in[0];
    const float* ori = (const float*)d_in[1];
    float* out = (float*)d_out;

    const int na = in_sizes[0] / 3;   // 16384
    const int nb = in_sizes[1] / 3;   // 16384

    // Zero the scalar accumulator (capture-legal async memset).
    hipMemsetAsync(d_out, 0, sizeof(float), stream);

    // 8 waves per block, 1 A-tile (16 points) per wave.
    dim3 block(256, 1, 1);
    dim3 grid_a((na / 16 + 7) / 8, 1, 1);
    dim3 grid_b((nb / 16 + 7) / 8, 1, 1);

    // adv -> ori direction
    chamfer_min_dir_kernel<<<grid_a, block, 0, stream>>>(
        adv, ori, nb, 1.0f / (float)na, out);
    // ori -> adv direction
    chamfer_min_dir_kernel<<<grid_b, block, 0, stream>>>(
        ori, adv, na, 1.0f / (float)nb, out);
}